// ShiftNetwork_79431125172453
// MI455X (gfx1250) — compile-verified
//
#include <hip/hip_runtime.h>
#include <hip/hip_bf16.h>

typedef __attribute__((ext_vector_type(2))) float v2f;
typedef __attribute__((ext_vector_type(8))) float v8f;

#define SR   10
#define TT   1024
#define NK   32
#define HID  256
#define RC   (TT * SR)      // 10240 columns (and rows) of X
#define PM1  (TT - 1)       // 1023 gathered rows

// ---------------- zero scratch ----------------
__global__ void k_zero(float* p, int n) {
    int i = blockIdx.x * blockDim.x + threadIdx.x;
    if (i < n) p[i] = 0.0f;
}

// ---------------- p = basis_X @ B ; shifts = trunc(p) ----------------
__global__ void k_p_shifts(const float* __restrict__ basis,
                           const float* __restrict__ B,
                           float* __restrict__ out_p,
                           int* __restrict__ shifts) {
    int t = blockIdx.x * blockDim.x + threadIdx.x;
    if (t >= TT) return;
    float acc = 0.0f;
#pragma unroll
    for (int k = 0; k < NK; ++k) acc += basis[t * NK + k] * B[k];
    out_p[t] = acc;
    if (t < PM1) shifts[t] = (int)truncf(acc);
}

// ---------------- column sums of the rolled row-gather of X ----------------
// Dominant cost: 1023 rows x 10240 cols x 4B = 42 MB streamed once.
// block x: 40 blocks of 256 threads -> one thread per output column
// block y: 8 row-chunks of ~128 rows -> float atomics to combine
__global__ void k_colsum(const float* __restrict__ X,
                         const int*   __restrict__ shifts,
                         float* __restrict__ colsum,
                         float* __restrict__ colcnt) {
    int j  = blockIdx.x * blockDim.x + threadIdx.x;   // column 0..RC-1
    int i0 = blockIdx.y * 128;
    int i1 = i0 + 128; if (i1 > PM1) i1 = PM1;
    float s = 0.0f, c = 0.0f;
    for (int i = i0; i < i1; ++i) {
        int sh  = shifts[i];
        int src = j - sh;
        src %= RC; if (src < 0) src += RC;
        // one-touch stream: non-temporal so X doesn't rinse L2
        float v = __builtin_nontemporal_load(&X[(size_t)(i * SR) * RC + src]);
        if (!(v != v)) { s += v; c += 1.0f; }          // NaN mask
    }
    atomicAdd(&colsum[j], s);
    atomicAdd(&colcnt[j], c);
}

// ---------------- mean -> ffill -> xs[0]=X[0,0] -> subsample by SR ----------------
__global__ void k_xs(const float* __restrict__ X,
                     const float* __restrict__ colsum,
                     const float* __restrict__ colcnt,
                     float* __restrict__ xs) {
    if (threadIdx.x != 0 || blockIdx.x != 0) return;
    float last = 0.0f;
    for (int j = 0; j < RC; ++j) {
        float cnt = colcnt[j];
        float v = (cnt > 0.0f) ? colsum[j] / cnt : __builtin_nanf("");
        if (j == 0) {
            last = v;
        } else {
            if (v != v) v = last;   // forward-fill
            last = v;
        }
        if (j % SR == 0) {
            int idx = j / SR;
            if (idx < TT) xs[idx] = (j == 0) ? X[0] : v;
        }
    }
}

// ---------------- layer 1: outer product + bias + relu (fp32) ----------------
__global__ void k_layer1(const float* __restrict__ xs,
                         const float* __restrict__ W1,
                         const float* __restrict__ b1,
                         float* __restrict__ h1) {
    int idx = blockIdx.x * blockDim.x + threadIdx.x;
    if (idx >= TT * HID) return;
    int t = idx >> 8, c = idx & 255;
    float v = xs[t] * W1[c] + b1[c];
    h1[idx] = fmaxf(v, 0.0f);
}

// ---------------- full-fp32 WMMA GEMM: Out = relu(A @ W + bias) ----------------
// M=1024, N=K=256. One 16x16 C-tile per wave via V_WMMA_F32_16X16X4_F32
// (K=4 per issue, 64 issues per tile). 8 waves/block, 128 blocks.
// Register layouts per CDNA5 ISA 7.12.2 (wave32):
//   A 16x4 f32 : lanes 0-15 row M=l, VGPR{0,1}=K{0,1}; lanes 16-31 K{2,3}
//   B 4x16 f32 : lanes 0-15 col N=l, VGPR{0,1}=K{0,1}; lanes 16-31 K{2,3}
//   C 16x16 f32: VGPR v -> row v + (lane>=16)*8, col = lane&15
__global__ __launch_bounds__(256) void k_gemm_relu(const float* __restrict__ A,
                                                   const float* __restrict__ W,
                                                   const float* __restrict__ bias,
                                                   float* __restrict__ Out) {
    int wave = threadIdx.x >> 5;
    int lane = threadIdx.x & 31;
    int tile = blockIdx.x * 8 + wave;     // 0..1023
    int tm = tile >> 4;                   // 0..63   (M tiles)
    int tn = tile & 15;                   // 0..15   (N tiles)
    int half = lane >> 4;                 // 0/1
    int l16  = lane & 15;

    int arow = tm * 16 + l16;             // A row for this lane
    int bcol = tn * 16 + l16;             // B/C column for this lane

    v8f c = {};
#pragma unroll 4
    for (int kk = 0; kk < HID; kk += 4) {
        int k0 = kk + half * 2;
        v2f a, b;
        a.x = A[arow * HID + k0];
        a.y = A[arow * HID + k0 + 1];
        b.x = W[(size_t)k0 * HID + bcol];
        b.y = W[(size_t)(k0 + 1) * HID + bcol];
        c = __builtin_amdgcn_wmma_f32_16x16x4_f32(
                /*neg_a=*/false, a, /*neg_b=*/false, b,
                /*c_mod=*/(short)0, c, /*reuse_a=*/false, /*reuse_b=*/false);
    }
#pragma unroll
    for (int v = 0; v < 8; ++v) {
        int row = tm * 16 + v + half * 8;
        Out[row * HID + bcol] = fmaxf(c[v] + bias[bcol], 0.0f);
    }
}

// ---------------- layer 4: y[t] = H[t,:] . W4 + b4 (one wave per row) ----------------
__global__ void k_layer4(const float* __restrict__ H,
                         const float* __restrict__ W4,
                         const float* __restrict__ b4,
                         float* __restrict__ y) {
    int wave = threadIdx.x >> 5, lane = threadIdx.x & 31;
    int t = blockIdx.x * (blockDim.x >> 5) + wave;
    if (t >= TT) return;
    float s = 0.0f;
#pragma unroll
    for (int c = lane; c < HID; c += 32) s += H[t * HID + c] * W4[c];
#pragma unroll
    for (int off = 16; off > 0; off >>= 1) s += __shfl_xor(s, off, 32);
    if (lane == 0) y[t] = s + b4[0];
}

extern "C" void kernel_launch(void* const* d_in, const int* in_sizes, int n_in,
                              void* d_out, int out_size, void* d_ws, size_t ws_size,
                              hipStream_t stream) {
    const float* X     = (const float*)d_in[0];
    const float* basis = (const float*)d_in[1];
    const float* B     = (const float*)d_in[2];
    const float* W1    = (const float*)d_in[3];
    const float* b1    = (const float*)d_in[4];
    const float* W2    = (const float*)d_in[5];
    const float* b2    = (const float*)d_in[6];
    const float* W3    = (const float*)d_in[7];
    const float* b3    = (const float*)d_in[8];
    const float* W4    = (const float*)d_in[9];
    const float* b4    = (const float*)d_in[10];

    float* out = (float*)d_out;   // [0:1024) = y, [1024:2048) = p

    char* w = (char*)d_ws;
    float* colsum = (float*)(w);                        // 40960 B
    float* colcnt = (float*)(w + 40960);                // 40960 B
    int*   shifts = (int*)  (w + 81920);                //  4096 B
    float* xs     = (float*)(w + 86016);                //  4096 B
    float* h1     = (float*)(w + 90112);                // 1 MB
    float* h2     = (float*)(w + 90112 + 1048576);      // 1 MB
    float* h3     = (float*)(w + 90112 + 2097152);      // 1 MB

    // zero colsum+colcnt (contiguous 2*10240 floats)
    k_zero<<<80, 256, 0, stream>>>(colsum, 2 * RC);
    // p + shifts (p written directly into d_out[1024:])
    k_p_shifts<<<4, 256, 0, stream>>>(basis, B, out + TT, shifts);
    // the 42 MB streaming pass: dominant cost (~1.8 us at 23.3 TB/s)
    dim3 gcs(RC / 256, 8);
    k_colsum<<<gcs, 256, 0, stream>>>(X, shifts, colsum, colcnt);
    // mean + ffill + subsample
    k_xs<<<1, 32, 0, stream>>>(X, colsum, colcnt, xs);
    // MLP (full fp32, WMMA f32 16x16x4)
    k_layer1<<<(TT * HID) / 256, 256, 0, stream>>>(xs, W1, b1, h1);
    k_gemm_relu<<<128, 256, 0, stream>>>(h1, W2, b2, h2);
    k_gemm_relu<<<128, 256, 0, stream>>>(h2, W3, b3, h3);
    k_layer4<<<128, 256, 0, stream>>>(h3, W4, b4, out);
}